// Router_19207093748098
// MI455X (gfx1250) — compile-verified
//
#include <hip/hip_runtime.h>
#include <hip/hip_bf16.h>
#include <math.h>

typedef __attribute__((ext_vector_type(2))) float v2f;
typedef __attribute__((ext_vector_type(4))) float v4f;
typedef __attribute__((ext_vector_type(8))) float v8f;

#define EMB 1024
#define NEXP 8

// ---------------------------------------------------------------------------
// Kernel 1: logits = x @ w_g.T via V_WMMA_F32_16X16X4_F32.
// Block = 256 threads = 8 waves; each wave owns one 16-token tile (128 tokens
// per block). B is staged in LDS *transposed and zero-padded to 16 experts*:
//   ldsB[k*16 + e] = (e < 8) ? w_g[e][k] : 0
// so every lane loads B unconditionally (no EXEC toggling around ds_loads,
// WMMA needs EXEC all-ones anyway) and the 32 lanes of a wave hit distinct
// LDS banks (banks e and 32+e) instead of the 16-way conflict of a row-major
// stride-1024 layout.
// A layout (16x4 f32, ISA 7.12.2): lanes 0-15 -> M=lane, K={0,1} in v0/v1;
// lanes 16-31 -> M=lane-16, K={2,3}. B (4x16) mirrored.
// K is split into two independent accumulator chains (c0/c1) to halve the
// WMMA RAW dependency chain and let the streams pipeline.
// ---------------------------------------------------------------------------
__global__ void gate_wmma(const float* __restrict__ x,
                          const float* __restrict__ wg,
                          float* __restrict__ logits) {
  __shared__ float ldsB[EMB * 16];             // 64 KB, transposed + padded
  const int tid = threadIdx.x;

  // Cooperative transposed fill: i = e*EMB + k (e-major -> coalesced reads).
  for (int i = tid; i < 16 * EMB; i += 256) {
    const int e = i >> 10;                     // expert
    const int k = i & (EMB - 1);
    ldsB[k * 16 + e] = (e < NEXP) ? wg[i] : 0.0f;
  }
  __syncthreads();

  const int wave = tid >> 5;
  const int lane = tid & 31;
  const int tile = blockIdx.x * 8 + wave;      // 16-token tile
  const int row  = lane & 15;                  // M index (and N index for B)
  const int khalf = (lane >> 4) * 2;           // 0 or 2
  const long rowBase = (long)(tile * 16 + row) * EMB;

  v8f c0 = {0.f, 0.f, 0.f, 0.f, 0.f, 0.f, 0.f, 0.f};
  v8f c1 = {0.f, 0.f, 0.f, 0.f, 0.f, 0.f, 0.f, 0.f};
  const int HALF = EMB / 2;

  for (int kb = 0; kb < HALF; kb += 4) {
    v2f a0, b0, a1, b1;
    const int k0 = kb + khalf;
    const int k1 = kb + HALF + khalf;
    a0.x = x[rowBase + k0];
    a0.y = x[rowBase + k0 + 1];
    b0.x = ldsB[k0 * 16 + row];
    b0.y = ldsB[(k0 + 1) * 16 + row];
    a1.x = x[rowBase + k1];
    a1.y = x[rowBase + k1 + 1];
    b1.x = ldsB[k1 * 16 + row];
    b1.y = ldsB[(k1 + 1) * 16 + row];
    c0 = __builtin_amdgcn_wmma_f32_16x16x4_f32(
        false, a0, false, b0, (short)0, c0, false, false);
    c1 = __builtin_amdgcn_wmma_f32_16x16x4_f32(
        false, a1, false, b1, (short)0, c1, false, false);
  }
  v8f c = c0 + c1;

  // C layout: VGPR v -> lanes 0-15: (M=v, N=lane); lanes 16-31: (M=v+8).
  const int n = lane & 15;                     // expert (N) index
  const int mOff = (lane >> 4) * 8;
  if (n < NEXP) {
    const int tokenBase = tile * 16 + mOff;
    #pragma unroll
    for (int v = 0; v < 8; ++v)
      logits[(long)(tokenBase + v) * NEXP + n] = c[v];
  }
}

// ---------------------------------------------------------------------------
// Kernel 2: per-token top-2 (+ pair softmax). Tie break = lowest index first,
// matching jax.lax.top_k.
// ---------------------------------------------------------------------------
__global__ void topk_softmax(const float* __restrict__ logits,
                             int* __restrict__ idx01,
                             float* __restrict__ p01, int N) {
  int n = blockIdx.x * blockDim.x + threadIdx.x;
  if (n >= N) return;
  const v4f* lg = (const v4f*)(logits + (long)n * NEXP);
  v4f lo = lg[0], hi = lg[1];
  float v[8] = {lo.x, lo.y, lo.z, lo.w, hi.x, hi.y, hi.z, hi.w};

  int i0 = 0; float v0 = v[0];
  #pragma unroll
  for (int e = 1; e < NEXP; ++e) if (v[e] > v0) { v0 = v[e]; i0 = e; }
  int i1 = -1; float v1 = -INFINITY;
  #pragma unroll
  for (int e = 0; e < NEXP; ++e)
    if (e != i0 && v[e] > v1) { v1 = v[e]; i1 = e; }

  float e1  = __expf(v1 - v0);                 // exp(max)=1 after shift
  float inv = 1.0f / (1.0f + e1);
  idx01[n] = i0; idx01[N + n] = i1;
  p01[n]  = inv; p01[N + n]  = e1 * inv;
}

// ---------------------------------------------------------------------------
// Kernel 3: capacity ranking. Single block of 256 threads; each thread owns a
// contiguous chunk of N/256 tokens. Per-k: LDS histogram -> per-expert
// exclusive scan over chunks -> replay for exact in-order ranks. k=1 ranks
// are offset by k=0 totals (priority order is k-major). Also emits
// used_capacity[e] = min(total[e], capacity) as float.
// ---------------------------------------------------------------------------
__global__ void rank_scan(const int* __restrict__ idx01,
                          int* __restrict__ rank01,
                          float* __restrict__ used_cap,
                          int N, int capacity) {
  __shared__ int hist[256 * NEXP];
  __shared__ int totals[NEXP];
  const int tid = threadIdx.x;
  const int chunk = N / 256;

  int kbase[NEXP];
  #pragma unroll
  for (int e = 0; e < NEXP; ++e) kbase[e] = 0;

  for (int k = 0; k < 2; ++k) {
    const int* idx = idx01 + k * N;
    int* rank = rank01 + k * N;

    int c[NEXP];
    #pragma unroll
    for (int e = 0; e < NEXP; ++e) c[e] = 0;
    for (int t = 0; t < chunk; ++t) c[idx[tid * chunk + t]]++;
    #pragma unroll
    for (int e = 0; e < NEXP; ++e) hist[tid * NEXP + e] = c[e];
    __syncthreads();

    if (tid < NEXP) {                          // exclusive scan over chunks
      int run = 0;
      for (int j = 0; j < 256; ++j) {
        int tmp = hist[j * NEXP + tid];
        hist[j * NEXP + tid] = run;
        run += tmp;
      }
      totals[tid] = run;
    }
    __syncthreads();

    int off[NEXP];
    #pragma unroll
    for (int e = 0; e < NEXP; ++e) off[e] = hist[tid * NEXP + e] + kbase[e];
    for (int t = 0; t < chunk; ++t) {
      int e = idx[tid * chunk + t];
      rank[tid * chunk + t] = off[e]++;
    }
    __syncthreads();

    if (k == 0) {
      #pragma unroll
      for (int e = 0; e < NEXP; ++e) kbase[e] = totals[e];
    } else if (tid < NEXP) {
      int total = kbase[tid] + totals[tid];
      used_cap[tid] = (float)(total < capacity ? total : capacity);
    }
    __syncthreads();
  }
}

// ---------------------------------------------------------------------------
// Kernel 4: fused zero-fill + scatter of cb_weight and sec_mask. One block per
// token streams its [8, capacity] slab of both outputs with non-temporal
// 128-bit stores, patching the <=2 nonzero entries in registers. This is the
// bandwidth-dominant pass (~1.34 GB of stores, ~58 us at 23.3 TB/s).
// ---------------------------------------------------------------------------
__global__ void fill_scatter(const int* __restrict__ idx01,
                             const int* __restrict__ rank01,
                             const float* __restrict__ p01,
                             float* __restrict__ cb,
                             float* __restrict__ mask,
                             int N, int capacity) {
  const int n = blockIdx.x;
  const int tid = threadIdx.x;
  const int slab = NEXP * capacity;

  const int i0 = idx01[n],     i1 = idx01[N + n];
  const int r0 = rank01[n],    r1 = rank01[N + n];
  const float q0 = p01[n],     q1 = p01[N + n];
  const int t0 = (r0 < capacity) ? i0 * capacity + r0 : -1;
  const int t1 = (r1 < capacity) ? i1 * capacity + r1 : -1;

  float* __restrict__ cbp = cb   + (long)n * slab;
  float* __restrict__ mkp = mask + (long)n * slab;

  for (int base = tid * 4; base < slab; base += blockDim.x * 4) {
    v4f w = {0.f, 0.f, 0.f, 0.f};
    v4f m = {0.f, 0.f, 0.f, 0.f};
    int d0 = t0 - base, d1 = t1 - base;
    if ((unsigned)d0 < 4u) { w[d0] = q0; m[d0] = 1.0f; }
    if ((unsigned)d1 < 4u) { w[d1] = q1; m[d1] = 1.0f; }
    __builtin_nontemporal_store(w, (v4f*)(cbp + base));
    __builtin_nontemporal_store(m, (v4f*)(mkp + base));
  }
}

// ---------------------------------------------------------------------------
extern "C" void kernel_launch(void* const* d_in, const int* in_sizes, int n_in,
                              void* d_out, int out_size, void* d_ws, size_t ws_size,
                              hipStream_t stream) {
  (void)n_in; (void)out_size; (void)ws_size;
  const float* x  = (const float*)d_in[0];   // [N, 1024]
  const float* wg = (const float*)d_in[1];   // [8, 1024]
  const int N = in_sizes[0] / EMB;           // 8192

  // capacity = floor(2 * 1.25 * N / 8), rounded up to even, min 4
  int capacity = (int)((2.0 * 1.25 * (double)N) / (double)NEXP);
  capacity += capacity & 1;
  if (capacity < 4) capacity = 4;            // 2560 for N=8192

  // Workspace partition (< 2 MB total).
  float* logits = (float*)d_ws;              // N*8 floats
  int*   idx01  = (int*)(logits + (size_t)N * NEXP);   // 2N ints
  int*   rank01 = idx01 + 2 * (size_t)N;               // 2N ints
  float* p01    = (float*)(rank01 + 2 * (size_t)N);    // 2N floats

  // Output layout: [used_capacity(8) | cb_weight(N*8*cap) | sec_mask(N*8*cap)]
  float* used = (float*)d_out;
  float* cb   = used + NEXP;
  float* mask = cb + (size_t)N * NEXP * capacity;

  gate_wmma   <<<N / 128, 256, 0, stream>>>(x, wg, logits);
  topk_softmax<<<(N + 255) / 256, 256, 0, stream>>>(logits, idx01, p01, N);
  rank_scan   <<<1, 256, 0, stream>>>(idx01, rank01, used, N, capacity);
  fill_scatter<<<N, 256, 0, stream>>>(idx01, rank01, p01, cb, mask, N, capacity);
}